// AttentiveRNN_7499012898916
// MI455X (gfx1250) — compile-verified
//
#include <hip/hip_runtime.h>
#include <math.h>

typedef float v2f __attribute__((ext_vector_type(2)));
typedef float v8f __attribute__((ext_vector_type(8)));
typedef int   v4i __attribute__((vector_size(16)));   // b128 payload type for async builtins

#define T_DIM 64
#define B_DIM 256
#define DIN_DIM 1024
#define H_DIM 512
#define K_DIM 64
#define A_DIM 4
#define W_DIM 16

// GEMM tiling
#define BM 128      // block rows (8 waves x 16)
#define BN 64       // block cols
#define BK 16       // K chunk
#define AST 20      // padded LDS row stride for A tile (20*4=80B: 16B aligned, conflict-free)

#if __has_builtin(__builtin_amdgcn_global_load_async_to_lds_b128) && \
    __has_builtin(__builtin_amdgcn_s_wait_asynccnt)
#define USE_ASYNC 1
#else
#define USE_ASYNC 0
#endif

#if USE_ASYNC
#define GAS(p) ((__attribute__((address_space(1))) v4i*)(p))
#define LAS(p) ((__attribute__((address_space(3))) v4i*)(p))
__device__ __forceinline__ void async_quad(const float* g, float* l) {
    __builtin_amdgcn_global_load_async_to_lds_b128(GAS(g), LAS(l), 0, 0);
}
#endif

// Generic fp32 WMMA GEMM: D = act( A[MxK] @ B[KxN] (+ bias[N]) (+ Cadd[MxN]) )
// Block = 256 threads = 8 waves; block tile 128x64, each wave 16x64.
// Requires M % 128 == 0, N % 64 == 0, K % 16 == 0 (true for all shapes used here).
__global__ __launch_bounds__(256) void gemm16_f32(
    const float* __restrict__ A, const float* __restrict__ Bm,
    const float* __restrict__ bias, const float* __restrict__ Cadd,
    float* __restrict__ D, int M, int N, int K, int do_relu)
{
    __shared__ float As[2][BM * AST];
    __shared__ float Bs[2][BK * BN];

    const int tid  = threadIdx.x;
    const int lane = tid & 31;
    const int wave = tid >> 5;
    const int rblk = blockIdx.y * BM;
    const int c0   = blockIdx.x * BN;
    const int mrow = lane & 15;            // tile row (A) / tile col (B)
    const int kh   = (lane >> 4) << 1;     // 0 or 2: K pair selected by lane half

    // Cooperative-load mapping (float4 granularity)
    const int aq_row = tid >> 2;           // 0..63   (+64 for second quad)
    const int aq_col = (tid & 3) << 2;     // 0,4,8,12
    const int bq_k   = tid >> 4;           // 0..15
    const int bq_n   = (tid & 15) << 2;    // 0..60

    const float* Ag0 = A  + (size_t)(rblk + aq_row)      * K + aq_col;
    const float* Ag1 = A  + (size_t)(rblk + aq_row + 64) * K + aq_col;
    const float* Bg  = Bm + (size_t)bq_k * N + c0 + bq_n;
    const size_t bstep = (size_t)BK * N;

    float* Al0[2] = { &As[0][aq_row * AST + aq_col],        &As[1][aq_row * AST + aq_col] };
    float* Al1[2] = { &As[0][(aq_row + 64) * AST + aq_col], &As[1][(aq_row + 64) * AST + aq_col] };
    float* Bl[2]  = { &Bs[0][bq_k * BN + bq_n],             &Bs[1][bq_k * BN + bq_n] };

    v8f acc[4] = {};
    const int nch = K / BK;
    int buf = 0;

#if USE_ASYNC
    // prime the pipeline: chunk 0 -> buffer 0
    async_quad(Ag0, Al0[0]);
    async_quad(Ag1, Al1[0]);
    async_quad(Bg,  Bl[0]);
#endif

    for (int c = 0; c < nch; ++c) {
#if USE_ASYNC
        __builtin_amdgcn_s_wait_asynccnt(0);   // chunk c resident in LDS
        __syncthreads();                        // visible to all waves; prev buffer free
        if (c + 1 < nch) {                      // prefetch chunk c+1 into other buffer
            const float* a0 = Ag0 + (size_t)(c + 1) * BK;
            const float* a1 = Ag1 + (size_t)(c + 1) * BK;
            const float* bg = Bg  + (size_t)(c + 1) * bstep;
            async_quad(a0, Al0[buf ^ 1]);
            async_quad(a1, Al1[buf ^ 1]);
            async_quad(bg, Bl[buf ^ 1]);
        }
#else
        __syncthreads();
        {   // synchronous staging fallback (single buffer)
            float4 qa0 = *(const float4*)(Ag0 + (size_t)c * BK);
            float4 qa1 = *(const float4*)(Ag1 + (size_t)c * BK);
            float4 qb  = *(const float4*)(Bg  + (size_t)c * bstep);
            *(float4*)Al0[0] = qa0;
            *(float4*)Al1[0] = qa1;
            *(float4*)Bl[0]  = qb;
        }
        __syncthreads();
#endif
        const float* Asb = As[buf];
        const float* Bsb = Bs[buf];
#pragma unroll
        for (int ks = 0; ks < 4; ++ks) {
            const int kb = ks * 4 + kh;
            v2f a;
            const float* ap = Asb + (wave * 16 + mrow) * AST + kb;
            a.x = ap[0];
            a.y = ap[1];
#pragma unroll
            for (int nt = 0; nt < 4; ++nt) {
                v2f b;
                b.x = Bsb[kb * BN + mrow + nt * 16];
                b.y = Bsb[(kb + 1) * BN + mrow + nt * 16];
                acc[nt] = __builtin_amdgcn_wmma_f32_16x16x4_f32(
                    false, a, false, b, (short)0, acc[nt], false, false);
            }
        }
#if USE_ASYNC
        buf ^= 1;
#endif
    }

    // C/D layout: VGPR r -> row r (lanes 0-15) or r+8 (lanes 16-31), col = lane%16
    const int rbase = rblk + wave * 16 + ((lane >> 4) << 3);
    const int cb    = c0 + mrow;
#pragma unroll
    for (int nt = 0; nt < 4; ++nt) {
        const int col = cb + nt * 16;
#pragma unroll
        for (int r = 0; r < 8; ++r) {
            const int row = rbase + r;
            float v = acc[nt][r];
            if (bias) v += bias[col];
            if (Cadd) v += Cadd[(size_t)row * N + col];
            if (do_relu) v = fmaxf(v, 0.0f);
            D[(size_t)row * N + col] = v;
        }
    }
}

// Hbuf[0] = broadcast(first_hidden), ctx0 = broadcast(first_hidden)
__global__ void init_state(const float* __restrict__ fh, float* __restrict__ Hbuf0,
                           float* __restrict__ ctx)
{
    int i = blockIdx.x * blockDim.x + threadIdx.x;   // b*H + h, i < B*H
    float v = fh[i & (H_DIM - 1)];
    Hbuf0[i] = v;
    ctx[i]   = v;
}

// HK[0][b][k] = b_kh[k] + first_hidden @ W_kh
__global__ void init_hk0(const float* __restrict__ fh, const float* __restrict__ W_kh,
                         const float* __restrict__ b_kh, float* __restrict__ HK0)
{
    int i = blockIdx.x * blockDim.x + threadIdx.x;   // b*K + k, i < B*K
    int k = i & (K_DIM - 1);
    float s = b_kh[k];
    for (int h = 0; h < H_DIM; ++h) s += fh[h] * W_kh[h * K_DIM + k];
    HK0[i] = s;
}

// Per-step attention: scores -> masked softmax over window -> ctx -> actions.
// One block (256 threads) per batch row. Deterministic LDS reductions (no atomics).
__global__ void attn_step(const float* __restrict__ Hbuf, const float* __restrict__ HK,
                          const float* __restrict__ KC, const float* __restrict__ query,
                          const float* __restrict__ W_act, const float* __restrict__ b_act,
                          float* __restrict__ ctx, float* __restrict__ out, int t)
{
    const int b   = blockIdx.x;
    const int tid = threadIdx.x;
    const int lo  = (t - 14) > 0 ? (t - 14) : 0;  // first valid Hbuf slot
    const int nw  = t + 2 - lo;                   // valid window length (<=16)

    __shared__ float part[256];
    __shared__ float sc[W_DIM];
    __shared__ float attnw[W_DIM];
    __shared__ float sctx[H_DIM];

    // --- scores: sc[u] = sum_k q[k] * tanh(kc[b,k] + hk[lo+u,b,k]) ---
    {
        const int u  = tid >> 4;          // window slot 0..15
        const int k0 = (tid & 15) * 4;    // 4 k's per thread
        float p = 0.0f;
        if (u < nw) {
            const float* kcp = KC + b * K_DIM;
            const float* hkp = HK + ((size_t)(lo + u) * B_DIM + b) * K_DIM;
#pragma unroll
            for (int j = 0; j < 4; ++j) {
                const int k = k0 + j;
                p += query[k] * tanhf(kcp[k] + hkp[k]);
            }
        }
        part[tid] = p;
    }
    __syncthreads();
    if (tid < W_DIM) {
        float s = 0.0f;
        for (int j = 0; j < 16; ++j) s += part[tid * 16 + j];
        sc[tid] = s;
    }
    __syncthreads();
    // --- softmax over valid slots (serial over <=16 elems: deterministic) ---
    if (tid == 0) {
        float m = -3.0e38f;
        for (int u = 0; u < nw; ++u) m = fmaxf(m, sc[u]);
        float s = 0.0f;
        for (int u = 0; u < nw; ++u) { float e = expf(sc[u] - m); attnw[u] = e; s += e; }
        float inv = 1.0f / s;
        for (int u = 0; u < nw; ++u) attnw[u] *= inv;
    }
    __syncthreads();
    // --- ctx[b,h] = sum_u attn[u] * Hbuf[lo+u,b,h] ---
    for (int h = tid; h < H_DIM; h += 256) {
        float acc = 0.0f;
        for (int u = 0; u < nw; ++u)
            acc += attnw[u] * Hbuf[((size_t)(lo + u) * B_DIM + b) * H_DIM + h];
        sctx[h] = acc;
        ctx[b * H_DIM + h] = acc;          // feeds next step's kc
    }
    __syncthreads();
    // --- actions[t,b,a] = ctx @ W_act + b_act ---
    {
        const int a  = tid & 3;
        const int h0 = tid >> 2;           // 0..63
        float p = 0.0f;
#pragma unroll
        for (int j = 0; j < 8; ++j) {
            const int h = h0 + 64 * j;
            p += sctx[h] * W_act[h * A_DIM + a];
        }
        part[tid] = p;
    }
    __syncthreads();
    if (tid < A_DIM) {
        float s = 0.0f;
        for (int g = 0; g < 64; ++g) s += part[tid + 4 * g];
        out[((size_t)t * B_DIM + b) * A_DIM + tid] = s + b_act[tid];
    }
}

extern "C" void kernel_launch(void* const* d_in, const int* in_sizes, int n_in,
                              void* d_out, int out_size, void* d_ws, size_t ws_size,
                              hipStream_t stream)
{
    (void)in_sizes; (void)n_in; (void)out_size; (void)ws_size;
    const float* x     = (const float*)d_in[0];
    const float* W_in  = (const float*)d_in[1];
    const float* b_in  = (const float*)d_in[2];
    const float* W_hid = (const float*)d_in[3];   // [2H, H]
    const float* b_hid = (const float*)d_in[4];
    const float* W_kc  = (const float*)d_in[5];
    const float* b_kc  = (const float*)d_in[6];
    const float* W_kh  = (const float*)d_in[7];
    const float* b_kh  = (const float*)d_in[8];
    const float* query = (const float*)d_in[9];
    const float* fh    = (const float*)d_in[10];
    const float* W_act = (const float*)d_in[11];
    const float* b_act = (const float*)d_in[12];
    float* out = (float*)d_out;                    // [T, B, A] fp32

    const size_t TB = (size_t)T_DIM * B_DIM;       // 16384
    float* XP   = (float*)d_ws;                                     // [T*B, H]
    float* Z    = XP   + TB * H_DIM;                                // [T*B, H]
    float* Hbuf = Z    + TB * H_DIM;                                // [T+1, B, H]
    float* HK   = Hbuf + (size_t)(T_DIM + 1) * B_DIM * H_DIM;       // [T+1, B, K]
    float* KC   = HK   + (size_t)(T_DIM + 1) * B_DIM * K_DIM;       // [B, K]
    float* ctx  = KC   + (size_t)B_DIM * K_DIM;                     // [B, H]

    dim3 blk(256);

    // Parallel precompute: XP = relu(x @ W_in + b_in)  [16384 x 1024 x 512]
    gemm16_f32<<<dim3(H_DIM / 64, TB / 128), blk, 0, stream>>>(
        x, W_in, b_in, nullptr, XP, (int)TB, H_DIM, DIN_DIM, 1);
    // Parallel precompute: Z = XP @ W_hid[H:2H] + b_hid  [16384 x 512 x 512]
    gemm16_f32<<<dim3(H_DIM / 64, TB / 128), blk, 0, stream>>>(
        XP, W_hid + (size_t)H_DIM * H_DIM, b_hid, nullptr, Z, (int)TB, H_DIM, H_DIM, 0);
    // Initial state
    init_state<<<(B_DIM * H_DIM) / 256, blk, 0, stream>>>(fh, Hbuf, ctx);
    init_hk0<<<(B_DIM * K_DIM) / 256, blk, 0, stream>>>(fh, W_kh, b_kh, HK);

    // Sequential recurrence over T
    for (int t = 0; t < T_DIM; ++t) {
        const float* Hprev = Hbuf + (size_t)t * B_DIM * H_DIM;
        float*       Hcur  = Hbuf + (size_t)(t + 1) * B_DIM * H_DIM;
        // h_t = relu(h_{t-1} @ W_hid[0:H] + Z_t)   [256 x 512 x 512]
        gemm16_f32<<<dim3(H_DIM / 64, B_DIM / 128), blk, 0, stream>>>(
            Hprev, W_hid, nullptr, Z + (size_t)t * B_DIM * H_DIM, Hcur,
            B_DIM, H_DIM, H_DIM, 1);
        // hk_t = h_t @ W_kh + b_kh   [256 x 512 x 64]
        gemm16_f32<<<dim3(K_DIM / 64, B_DIM / 128), blk, 0, stream>>>(
            Hcur, W_kh, b_kh, nullptr, HK + (size_t)(t + 1) * B_DIM * K_DIM,
            B_DIM, K_DIM, H_DIM, 0);
        // kc_t = ctx_{t-1} @ W_kc + b_kc   [256 x 512 x 64]
        gemm16_f32<<<dim3(K_DIM / 64, B_DIM / 128), blk, 0, stream>>>(
            ctx, W_kc, b_kc, nullptr, KC, B_DIM, K_DIM, H_DIM, 0);
        // attention + ctx update + actions
        attn_step<<<B_DIM, blk, 0, stream>>>(Hbuf, HK, KC, query, W_act, b_act,
                                             ctx, out, t);
    }
}